// Dot_Attention_4913442586814
// MI455X (gfx1250) — compile-verified
//
#include <hip/hip_runtime.h>
#include <math.h>

// Problem constants (from reference): M=2048, S=512, D=384, N_SENT=512.
#define M_ROWS 2048
#define S_LEN  512
#define D_DIM  384
#define NSENT  512

typedef __attribute__((ext_vector_type(2))) float v2f;
typedef __attribute__((ext_vector_type(8))) float v8f;

// ---------------------------------------------------------------------------
// Kernel 1: exclusive prefix sum of lengths[] (Hillis-Steele scan, 1 block).
// prefix[i] = sum_{j<i} lengths[j]
// ---------------------------------------------------------------------------
__global__ __launch_bounds__(NSENT) void prefix_scan_kernel(
    const int* __restrict__ lengths, int* __restrict__ prefix) {
  __shared__ int buf[NSENT];
  const int t = threadIdx.x;
  buf[t] = lengths[t];
  __syncthreads();
  for (int off = 1; off < NSENT; off <<= 1) {
    int v = (t >= off) ? buf[t - off] : 0;
    __syncthreads();
    buf[t] += v;
    __syncthreads();
  }
  prefix[t] = (t == 0) ? 0 : buf[t - 1];
}

// ---------------------------------------------------------------------------
// Kernel 2: one block (8 wave32 = 256 threads) per query row m.
// Each wave owns 4 s-tiles of 16 rows, computed with V_WMMA_F32_16X16X4_F32:
//   A (16x4, documented layout) = q chunk replicated in all 16 M-rows
//   B (4x16)                    = 16 V rows (lane = column = s row)
//   => every D column j = dot(q, V[s_base+j]); read row 0 (acc[0]).
// Fully-masked tiles are skipped (no V traffic) -> ~25% HBM savings.
// ---------------------------------------------------------------------------
__global__ __launch_bounds__(256) void attn_softmax_kernel(
    const float* __restrict__ queries,   // [M, D]
    const float* __restrict__ values,    // [M, S, D]
    const int*   __restrict__ idx,       // [M, 5]
    const int*   __restrict__ lengths,   // [NSENT]
    const int*   __restrict__ prefix,    // [NSENT]  (exclusive cumsum)
    float*       __restrict__ out)       // [M, S]
{
  const int m      = blockIdx.x;
  const int tid    = threadIdx.x;
  const int lane   = tid & 31;
  const int wave   = tid >> 5;    // 0..7
  const int half   = lane >> 4;   // 0 or 1  (K-pair selector)
  const int lane16 = lane & 15;   // s row within tile / N column

  __shared__ __align__(16) float q_lds[D_DIM];
  __shared__ float alpha[S_LEN];
  __shared__ float red_max[8];
  __shared__ float red_sum[8];

  // Stage q[m] into LDS (read once; A-side reads broadcast from here).
  for (int i = tid; i < D_DIM; i += 256)
    q_lds[i] = queries[(size_t)m * D_DIM + i];

  // Scalar (block-uniform) mask parameters.
  const int sent_id = idx[m * 5 + 4];
  const int mlen    = lengths[sent_id];          // mask_a: s >= mlen invalid
  const int total_l = prefix[sent_id];
  const int start   = idx[m * 5 + 2] - total_l;  // mask_b: [start, end) invalid
  const int end     = idx[m * 5 + 3] - total_l;
  __syncthreads();

  const float sc = 1.0f / sqrtf((float)D_DIM);

  // ---- score computation: 4 tiles per wave, interleaved across waves ----
  for (int i = 0; i < 4; ++i) {
    const int tile   = wave + 8 * i;     // 0..31
    const int s_base = tile * 16;
    const int s      = s_base + lane16;

    // Whole-tile skip (wave-uniform): saves all V bandwidth for this tile.
    const bool skip_all =
        (s_base >= mlen) || ((s_base >= start) && (s_base + 16 <= end));
    if (skip_all) {
      if (half == 0) alpha[s] = -__builtin_inff();
      continue;
    }

    v8f acc = {};
    // B fragment: lane -> V row (s_base + lane16), two K elems per half-wave.
    const float* __restrict__ vrow =
        values + ((size_t)m * S_LEN + s_base + lane16) * D_DIM + 2 * half;
    const int qoff = 2 * half;

#pragma unroll 4
    for (int d0 = 0; d0 < D_DIM; d0 += 4) {
      // A fragment: q chunk, identical in both half-wave K slots -> every
      // A row equals q[d0..d0+3]. LDS same-address reads broadcast.
      v2f a = *(const v2f*)(&q_lds[d0 + qoff]);
      // B fragment: global_load_b64 of V[s, d0+2*half .. +1].
      v2f b = *(const v2f*)(vrow + d0);
      acc = __builtin_amdgcn_wmma_f32_16x16x4_f32(
          /*neg_a=*/false, a, /*neg_b=*/false, b,
          /*c_mod=*/(short)0, acc, /*reuse_a=*/false, /*reuse_b=*/false);
    }

    // Row 0 of D: lanes 0-15 hold alpha[s_base + lane] in acc[0].
    if (half == 0) {
      const bool valid = (s < mlen) && !((s >= start) && (s < end));
      alpha[s] = valid ? acc[0] * sc : -__builtin_inff();
    }
  }
  __syncthreads();

  // ---- block softmax over S=512 (2 elements per thread) ----
  const float v0 = alpha[tid];
  const float v1 = alpha[tid + 256];

  float mx = fmaxf(v0, v1);
  for (int off = 16; off > 0; off >>= 1)
    mx = fmaxf(mx, __shfl_xor(mx, off, 32));
  if (lane == 0) red_max[wave] = mx;
  __syncthreads();
  if (tid == 0) {
    float r = red_max[0];
    for (int w = 1; w < 8; ++w) r = fmaxf(r, red_max[w]);
    red_max[0] = r;
  }
  __syncthreads();
  mx = red_max[0];

  const float e0 = expf(v0 - mx);
  const float e1 = expf(v1 - mx);
  float sum = e0 + e1;
  for (int off = 16; off > 0; off >>= 1)
    sum += __shfl_xor(sum, off, 32);
  if (lane == 0) red_sum[wave] = sum;
  __syncthreads();
  if (tid == 0) {
    float r = 0.0f;
    for (int w = 0; w < 8; ++w) r += red_sum[w];
    red_sum[0] = r;
  }
  __syncthreads();
  const float inv = 1.0f / red_sum[0];

  out[(size_t)m * S_LEN + tid]       = e0 * inv;
  out[(size_t)m * S_LEN + tid + 256] = e1 * inv;
}

// ---------------------------------------------------------------------------
// Launch wrapper. Inputs (setup_inputs order): queries f32[M*D],
// values f32[M*S*D], idx int[M*5], lengths int[NSENT]. Output f32[M*S].
// Workspace: prefix sums (NSENT ints).
// ---------------------------------------------------------------------------
extern "C" void kernel_launch(void* const* d_in, const int* in_sizes, int n_in,
                              void* d_out, int out_size, void* d_ws, size_t ws_size,
                              hipStream_t stream) {
  (void)in_sizes; (void)n_in; (void)out_size; (void)ws_size;
  const float* queries = (const float*)d_in[0];
  const float* values  = (const float*)d_in[1];
  const int*   idx     = (const int*)d_in[2];
  const int*   lengths = (const int*)d_in[3];
  int* prefix = (int*)d_ws;

  prefix_scan_kernel<<<1, NSENT, 0, stream>>>(lengths, prefix);
  attn_softmax_kernel<<<M_ROWS, 256, 0, stream>>>(
      queries, values, idx, lengths, prefix, (float*)d_out);
}